// GraphConvLayer_18657337934720
// MI455X (gfx1250) — compile-verified
//
#include <hip/hip_runtime.h>

typedef __attribute__((ext_vector_type(2))) float v2f;
typedef __attribute__((ext_vector_type(8))) float v8f;

#define NNODES 100000
#define DIM    128

// ---------------------------------------------------------------------------
// Kernel 1: support = features @ W  via V_WMMA_F32_16X16X4_F32
// Block = 128 threads = 4 wave32; each wave computes a 16x128 output strip.
// W staged in LDS once per block, K-PAIR INTERLEAVED so each lane's B
// fragment {W[kk][c], W[kk+1][c]} is one aligned ds_load_b64.
//   sW[((k>>1)*DIM + c)*2 + (k&1)] = W[k][c]
// ---------------------------------------------------------------------------
__global__ __launch_bounds__(128) void gcn_gemm_wmma(
    const float* __restrict__ F, const float* __restrict__ W,
    float* __restrict__ S) {
  __shared__ float sW[DIM * DIM];  // 64 KB, interleaved layout

  // Cooperative load of W into LDS (float4 global reads, interleaved scatter)
  {
    const float4* Wv = (const float4*)W;
    int t = threadIdx.x;
#pragma unroll
    for (int i = 0; i < (DIM * DIM / 4) / 128; ++i) {
      int idx = t + i * 128;        // float4 index
      int r   = idx >> 5;           // row (DIM/4 = 32 float4 per row)
      int c   = (idx & 31) * 4;     // starting column
      float4 w = Wv[idx];
      int base = ((r >> 1) * DIM + c) * 2 + (r & 1);
      sW[base + 0] = w.x;
      sW[base + 2] = w.y;
      sW[base + 4] = w.z;
      sW[base + 6] = w.w;
    }
  }
  __syncthreads();

  const int lane = threadIdx.x & 31;
  const int wave = threadIdx.x >> 5;
  const int rowBase = blockIdx.x * 64 + wave * 16;
  if (rowBase >= NNODES) return;  // NNODES % 16 == 0: whole-tile guard

  const int m     = rowBase + (lane & 15);  // A row this lane feeds
  const int half  = lane >> 4;              // lanes 16-31 hold K+2, K+3
  const int c     = lane & 15;              // B/D column within tile

  v8f acc[8] = {};  // 8 N-tiles of 16 cols -> 16x128 strip

  const float2* Fv = (const float2*)F;  // rows 512B-aligned, kk even
  for (int k = 0; k < DIM; k += 4) {
    const int kp = (k >> 1) + half;  // K-pair index this half-wave consumes
    float2 af = Fv[(size_t)m * (DIM / 2) + kp];
    v2f a;
    a.x = af.x;  // A[m][2*kp]
    a.y = af.y;  // A[m][2*kp+1]
    const v2f* sWp = (const v2f*)&sW[kp * DIM * 2];  // this K-pair's 128 cols
#pragma unroll
    for (int j = 0; j < 8; ++j) {
      v2f b = sWp[16 * j + c];  // single aligned ds_load_b64
      acc[j] = __builtin_amdgcn_wmma_f32_16x16x4_f32(
          /*neg_a=*/false, a, /*neg_b=*/false, b,
          /*c_mod=*/(short)0, acc[j], /*reuse_a=*/false, /*reuse_b=*/false);
    }
  }

  // D layout: VGPR r -> row rowBase+r (lanes 0-15) / rowBase+r+8 (lanes 16-31)
  const int r0 = rowBase + (half << 3);
#pragma unroll
  for (int j = 0; j < 8; ++j) {
#pragma unroll
    for (int r = 0; r < 8; ++r) {
      S[(size_t)(r0 + r) * DIM + 16 * j + c] = acc[j][r];
    }
  }
}

// ---------------------------------------------------------------------------
// Kernel 2: zero-fill accumulator (d_out is poisoned by the harness)
// ---------------------------------------------------------------------------
__global__ void gcn_zero(float4* __restrict__ p, int n4) {
  int i = blockIdx.x * blockDim.x + threadIdx.x;
  if (i < n4) p[i] = make_float4(0.f, 0.f, 0.f, 0.f);
}

// ---------------------------------------------------------------------------
// Kernel 3: edge scatter  out[dst] += vals[e] * support[src]
// One wave32 per edge; lane owns float4 -> 512B coalesced gather +
// 128 contiguous global_atomic_add_f32 (L2-resident accumulator).
// ---------------------------------------------------------------------------
__global__ __launch_bounds__(256) void gcn_scatter(
    const float* __restrict__ S, const int* __restrict__ src,
    const int* __restrict__ dst, const float* __restrict__ vals,
    float* __restrict__ out, int nE) {
  int gid  = blockIdx.x * blockDim.x + threadIdx.x;
  int e    = gid >> 5;
  int lane = gid & 31;
  if (e >= nE) return;

  int   s = src[e];
  int   d = dst[e];
  float v = vals[e];

  float4 m = ((const float4*)(S + (size_t)s * DIM))[lane];
  float* drow = out + (size_t)d * DIM + lane * 4;
  unsafeAtomicAdd(drow + 0, m.x * v);
  unsafeAtomicAdd(drow + 1, m.y * v);
  unsafeAtomicAdd(drow + 2, m.z * v);
  unsafeAtomicAdd(drow + 3, m.w * v);
}

// ---------------------------------------------------------------------------
// Kernel 4: out = relu(out + b) + features   (in-place, float4 streams)
// ---------------------------------------------------------------------------
__global__ void gcn_finalize(float4* __restrict__ out,
                             const float4* __restrict__ feat,
                             const float4* __restrict__ bias, int n4) {
  int i = blockIdx.x * blockDim.x + threadIdx.x;
  if (i >= n4) return;
  float4 o = out[i];
  float4 f = feat[i];
  float4 bb = bias[i & 31];  // DIM/4 == 32 float4 per row
  float4 r;
  r.x = fmaxf(o.x + bb.x, 0.f) + f.x;
  r.y = fmaxf(o.y + bb.y, 0.f) + f.y;
  r.z = fmaxf(o.z + bb.z, 0.f) + f.z;
  r.w = fmaxf(o.w + bb.w, 0.f) + f.w;
  out[i] = r;
}

// ---------------------------------------------------------------------------
extern "C" void kernel_launch(void* const* d_in, const int* in_sizes, int n_in,
                              void* d_out, int out_size, void* d_ws,
                              size_t ws_size, hipStream_t stream) {
  const float* features = (const float*)d_in[0];
  const int*   edge_src = (const int*)d_in[1];
  const int*   edge_dst = (const int*)d_in[2];
  const float* edge_val = (const float*)d_in[3];
  const float* W        = (const float*)d_in[4];
  const float* b        = (const float*)d_in[5];
  float* out     = (float*)d_out;
  float* support = (float*)d_ws;  // NNODES*DIM floats = 51.2 MB scratch

  const int nE = in_sizes[1];
  const int n4 = NNODES * DIM / 4;  // 3.2M float4

  // 1) support = features @ W  (WMMA f32)
  dim3 gGemm((NNODES + 63) / 64);
  gcn_gemm_wmma<<<gGemm, 128, 0, stream>>>(features, W, support);

  // 2) zero accumulator
  gcn_zero<<<(n4 + 255) / 256, 256, 0, stream>>>((float4*)out, n4);

  // 3) edge scatter-add (wave per edge)
  long long threads = (long long)nE * 32;
  gcn_scatter<<<(unsigned)((threads + 255) / 256), 256, 0, stream>>>(
      support, edge_src, edge_dst, edge_val, out, nE);

  // 4) bias + relu + residual
  gcn_finalize<<<(n4 + 255) / 256, 256, 0, stream>>>(
      (float4*)out, (const float4*)features, (const float4*)b, n4);
}